// LISTA_Model_41506563949091
// MI455X (gfx1250) — compile-verified
//
#include <hip/hip_runtime.h>

typedef __attribute__((ext_vector_type(2))) float v2f;
typedef __attribute__((ext_vector_type(8))) float v8f;

#define T_DIM 8192
#define N_DIM 512
#define M_DIM 2048
#define K_DIM 3
#define RHO_F 1e-4f
#define SCAN_BLOCKS 64

// =====================================================================
// Kernel 1:  Bs[k] = W_k @ W_k^T   (K x M x M), fp32 WMMA 16x16x4
// One wave32 per 16x16 output tile; loop over N=512 in K-chunks of 4.
// fp32 A-operand layout (16x4): lane L holds float2 of row (base+(L&15))
// at columns n + 2*(L>>4) .. +1.  B-operand (4x16) of W^T has the exact
// same per-lane load pattern on row-major W, so both feeds are b64 loads.
// =====================================================================
__global__ __launch_bounds__(256) void lista_compute_B(
    const float* __restrict__ A, float* __restrict__ Bs)
{
    const int wave  = threadIdx.x >> 5;
    const int lane  = threadIdx.x & 31;
    const int lrow  = lane & 15;
    const int lpair = lane >> 4;           // 0 or 1

    const int tilesPerK = (M_DIM / 16) * (M_DIM / 16);
    int tile = blockIdx.x * 8 + wave;      // 0 .. K*128*128-1
    int k  = tile / tilesPerK;
    int r  = tile % tilesPerK;
    int ti = r / (M_DIM / 16);
    int tj = r % (M_DIM / 16);

    const float* W  = A + (size_t)k * M_DIM * N_DIM;
    const float* pa = W + (size_t)(ti * 16 + lrow) * N_DIM + 2 * lpair;
    const float* pb = W + (size_t)(tj * 16 + lrow) * N_DIM + 2 * lpair;

    v8f c = {};
#pragma unroll 8
    for (int n = 0; n < N_DIM; n += 4) {
        v2f a = *(const v2f*)(pa + n);
        v2f b = *(const v2f*)(pb + n);
        // D = A(16x4) x B(4x16) + C   -> v_wmma_f32_16x16x4_f32
        c = __builtin_amdgcn_wmma_f32_16x16x4_f32(
                false, a, false, b, (short)0, c, false, false);
    }

    float* out = Bs + (size_t)k * M_DIM * M_DIM;
#pragma unroll
    for (int v = 0; v < 8; ++v) {
        int row = ti * 16 + v + 8 * lpair;   // C/D layout: M = v + 8*(lane>=16)
        int col = tj * 16 + lrow;            //             N = lane & 15
        out[(size_t)row * M_DIM + col] = c[v];
    }
}

// =====================================================================
// Kernel 2:  Xp[t,k,m] = sum_n x[t,n] * W_k[m,n]   (T x K x M)
// Same WMMA tiling: A-operand rows come from x, B-operand from W_k.
// =====================================================================
__global__ __launch_bounds__(256) void lista_compute_Xp(
    const float* __restrict__ x, const float* __restrict__ A,
    float* __restrict__ Xp)
{
    const int wave  = threadIdx.x >> 5;
    const int lane  = threadIdx.x & 31;
    const int lrow  = lane & 15;
    const int lpair = lane >> 4;

    const int tilesPerK = (T_DIM / 16) * (M_DIM / 16);
    int tile = blockIdx.x * 8 + wave;
    int k  = tile / tilesPerK;
    int r  = tile % tilesPerK;
    int tt = r / (M_DIM / 16);
    int tm = r % (M_DIM / 16);

    const float* W  = A + (size_t)k * M_DIM * N_DIM;
    const float* pa = x + (size_t)(tt * 16 + lrow) * N_DIM + 2 * lpair;
    const float* pb = W + (size_t)(tm * 16 + lrow) * N_DIM + 2 * lpair;

    v8f c = {};
#pragma unroll 8
    for (int n = 0; n < N_DIM; n += 4) {
        v2f a = *(const v2f*)(pa + n);
        v2f b = *(const v2f*)(pb + n);
        c = __builtin_amdgcn_wmma_f32_16x16x4_f32(
                false, a, false, b, (short)0, c, false, false);
    }

#pragma unroll
    for (int v = 0; v < 8; ++v) {
        int t = tt * 16 + v + 8 * lpair;
        int m = tm * 16 + lrow;
        Xp[((size_t)t * K_DIM + k) * M_DIM + m] = c[v];
    }
}

// =====================================================================
// Init: copy h0 into ping-pong buffer 0, zero the barrier counters.
// (harness poisons d_ws with 0xAA; must be re-done every launch)
// =====================================================================
__global__ void lista_init(const float* __restrict__ h0,
                           float* __restrict__ hbuf, unsigned* __restrict__ bar)
{
    int i = blockIdx.x * blockDim.x + threadIdx.x;
    if (i < M_DIM) hbuf[i] = h0[i];
    if (i < 2) bar[i] = 0u;
}

// ---------------------------------------------------------------------
// CDNA5 async global->LDS copy (16B per lane), tracked by ASYNCcnt.
// lds_off is a 32-bit LDS byte address (low 32 bits of the generic
// pointer to __shared__, per the ISA LDS-aperture truncation rule).
// ---------------------------------------------------------------------
__device__ __forceinline__ void async_copy_b128(unsigned lds_off,
                                                const float* gaddr)
{
    asm volatile("global_load_async_to_lds_b128 %0, %1, off"
                 :: "v"(lds_off), "v"(gaddr) : "memory");
}
__device__ __forceinline__ void wait_asynccnt0()
{
    asm volatile("s_wait_asynccnt 0" ::: "memory");
}

// ---------------------------------------------------------------------
// Grid-wide barrier: arrival counter + generation counter in d_ws.
// gen snapshot MUST precede the arrival atomicAdd.
// ---------------------------------------------------------------------
__device__ __forceinline__ void grid_barrier(unsigned* bar)
{
    __syncthreads();
    __threadfence();
    if (threadIdx.x == 0) {
        volatile unsigned* vgen = bar + 1;
        unsigned gen = *vgen;                         // snapshot first
        if (atomicAdd(bar, 1u) == SCAN_BLOCKS - 1u) { // last to arrive
            bar[0] = 0u;
            __threadfence();
            atomicAdd(bar + 1, 1u);                   // release next phase
        } else {
            while (*vgen == gen) __builtin_amdgcn_s_sleep(1);
        }
    }
    __syncthreads();
    __threadfence();
}

// =====================================================================
// Kernel 3: the sequential LISTA scan. 64 persistent blocks x 256 thr.
// Per (t,k): h DMA'd to LDS (async-to-LDS path), each wave32 owns 4 of
// this block's 32 rows; lane c accumulates float4 chunks of
// B[row, :] * h[:] (global_load_b128 / ds_load_b128), shfl_xor reduce,
// lane 0 applies the closed-form shrink:  h' = max(u - rho/a, 0).
// Bs (50 MB) is L2-resident on MI455X (192 MB L2) -> per-step reads of
// B_k stream from L2, not HBM.
// =====================================================================
__global__ __launch_bounds__(256) void lista_scan(
    const float* __restrict__ alpha,
    const float* __restrict__ Bs,
    const float* __restrict__ Xp,
    float* __restrict__ hbuf,       // 2 * M_DIM floats, ping-pong
    unsigned* __restrict__ bar,
    float* __restrict__ out)        // T x M
{
    __shared__ __align__(16) float hl[M_DIM];

    const int wave = threadIdx.x >> 5;
    const int lane = threadIdx.x & 31;
    const int rows_per_block = M_DIM / SCAN_BLOCKS;      // 32
    const int row0 = blockIdx.x * rows_per_block + wave * (rows_per_block / 8);

    const unsigned ldsbase = (unsigned)(uintptr_t)(void*)hl;

    // hoist the 3 alphas
    float a_k[K_DIM];
#pragma unroll
    for (int k = 0; k < K_DIM; ++k) a_k[k] = alpha[k + 1];

    for (int t = 0; t < T_DIM; ++t) {
        for (int k = 0; k < K_DIM; ++k) {
            const int it = t * K_DIM + k;
            const float* hsrc = hbuf + (size_t)(it & 1) * M_DIM;
            float*       hdst = hbuf + (size_t)((it + 1) & 1) * M_DIM;

            // stage h into LDS: 2048 floats = 512 x b128, async DMA
            {
                const float* g = hsrc + threadIdx.x * 4;
                unsigned     l = ldsbase + threadIdx.x * 16u;
                async_copy_b128(l,          g);            // floats [0,1024)
                async_copy_b128(l + 4096u,  g + 1024);     // floats [1024,2048)
                wait_asynccnt0();
            }
            __syncthreads();

            const float inva = 1.0f / a_k[k];
            const float thr  = RHO_F * inva;
            const float* B   = Bs + (size_t)k * M_DIM * M_DIM;
            const float* xp  = Xp + ((size_t)t * K_DIM + k) * M_DIM;
            const float4* hl4 = (const float4*)hl;

#pragma unroll
            for (int rr = 0; rr < rows_per_block / 8; ++rr) {
                const int row = row0 + rr;
                const float4* Brow4 = (const float4*)(B + (size_t)row * M_DIM);
                float s = 0.0f;
#pragma unroll 4
                for (int c = lane; c < M_DIM / 4; c += 32) {
                    float4 b4 = Brow4[c];
                    float4 h4 = hl4[c];
                    s = fmaf(b4.x, h4.x,
                        fmaf(b4.y, h4.y,
                        fmaf(b4.z, h4.z,
                        fmaf(b4.w, h4.w, s))));
                }
                // wave32 reduction
#pragma unroll
                for (int off = 16; off > 0; off >>= 1)
                    s += __shfl_xor(s, off, 32);
                if (lane == 0) {
                    float u = hl[row] - s * inva + xp[row] * inva;
                    float v = fmaxf(u - thr, 0.0f);
                    hdst[row] = v;
                    if (k == K_DIM - 1) out[(size_t)t * M_DIM + row] = v;
                }
            }
            grid_barrier(bar);
        }
    }
}

extern "C" void kernel_launch(void* const* d_in, const int* in_sizes, int n_in,
                              void* d_out, int out_size, void* d_ws, size_t ws_size,
                              hipStream_t stream)
{
    const float* x     = (const float*)d_in[0];   // [T, N]
    const float* A     = (const float*)d_in[1];   // [K, M, N]
    const float* alpha = (const float*)d_in[2];   // [K+1]
    const float* h0    = (const float*)d_in[3];   // [M]
    float* out = (float*)d_out;                   // [T, M]

    // workspace layout
    float* Bs   = (float*)d_ws;                                   // K*M*M
    float* Xp   = Bs + (size_t)K_DIM * M_DIM * M_DIM;             // T*K*M
    float* hbuf = Xp + (size_t)T_DIM * K_DIM * M_DIM;             // 2*M
    unsigned* bar = (unsigned*)(hbuf + 2 * M_DIM);                // 2

    // Phase 1: Bs = W W^T  (K*128*128 tiles, 8 waves/block)
    lista_compute_B<<<(K_DIM * (M_DIM/16) * (M_DIM/16)) / 8, 256, 0, stream>>>(A, Bs);
    // Phase 2: Xp = x W^T  (K*512*128 tiles)
    lista_compute_Xp<<<(K_DIM * (T_DIM/16) * (M_DIM/16)) / 8, 256, 0, stream>>>(x, A, Xp);
    // Phase 3: init + persistent scan
    lista_init<<<(M_DIM + 255) / 256, 256, 0, stream>>>(h0, hbuf, bar);
    lista_scan<<<SCAN_BLOCKS, 256, 0, stream>>>(alpha, Bs, Xp, hbuf, bar, out);
}